// GumbelSoftmaxFreeFormRNN_24300924961010
// MI455X (gfx1250) — compile-verified
//
#include <hip/hip_runtime.h>
#include <hip/hip_bf16.h>

typedef float v2f __attribute__((ext_vector_type(2)));
typedef float v8f __attribute__((ext_vector_type(8)));

#define RNN_B 32
#define RNN_T 512
#define RNN_I 256
#define RNN_N 1024
#define RNN_M 16
#define E_IN 8192
#define E_REC 8192
#define N_TOTAL 17408
#define RELU_CAP 1000000.0f
#define HS_STRIDE 1028   // 32 rows * 1028 floats; 1028%64==4 -> conflict-free b64 A reads, 16B aligned

// workspace layout (float offsets)
#define OFF_W4   0u        // packed W_rec: [k0/4][n][4]  -> 1024*1024
#define OFF_WIN  1048576u  // W_in_dense [i][n]           -> 256*1024
#define OFF_BIAS 1310720u  // 1024
#define OFF_WALL 1311744u  // all_weights 17408
#define OFF_H0   1329152u  // 32*1024
#define OFF_H1   1361920u  // 32*1024
#define WS_FLOATS 1394688u

// ---------------- zero workspace ----------------
__global__ void rnn_zero_ws(float4* ws4, unsigned count4) {
    unsigned i = blockIdx.x * blockDim.x + threadIdx.x;
    unsigned stride = gridDim.x * blockDim.x;
    float4 z = {0.f, 0.f, 0.f, 0.f};
    for (; i < count4; i += stride) ws4[i] = z;
}

// ---------------- all_weights[e] = grid[argmax(logits[e,:])] ----------------
__global__ void rnn_argmax_weights(const float* __restrict__ logits,
                                   const float* __restrict__ grid,
                                   float* __restrict__ wall) {
    int e = blockIdx.x * blockDim.x + threadIdx.x;
    if (e >= N_TOTAL) return;
    const float* row = logits + e * RNN_M;
    float best = row[0];
    int bi = 0;
#pragma unroll
    for (int j = 1; j < RNN_M; ++j) {
        float v = row[j];
        if (v > best) { best = v; bi = j; }   // strict '>' keeps first max (jnp.argmax tiebreak)
    }
    wall[e] = grid[bi];
}

// ---------------- scatter sparse edges into dense tables ----------------
// Thread n owns neuron n: its column of W_in and its row-block of packed W_rec -> race free.
__global__ void rnn_build_tables(const int* __restrict__ in_src,
                                 const int* __restrict__ rec_src,
                                 const float* __restrict__ wall,
                                 float* __restrict__ W4,    // [k0/4][n][4]
                                 float* __restrict__ Win,   // [i][n]
                                 float* __restrict__ bias) {
    int n = blockIdx.x * blockDim.x + threadIdx.x;
    if (n >= RNN_N) return;
    bias[n] = wall[E_IN + E_REC + n];
#pragma unroll
    for (int k = 0; k < 8; ++k) {
        int e = n * 8 + k;
        int i = in_src[e];
        Win[i * RNN_N + n] += wall[e];            // input one-hot -> row table
        int j = rec_src[e];
        W4[(unsigned)(j >> 2) * (RNN_N * 4) + n * 4 + (j & 3)] += wall[E_IN + e];
    }
}

// ---------------- one recurrent timestep: H_next = act(H_prev * W + Win[x_t] + b) ----------------
// 16 blocks x 128 threads (4 waves). Wave w handles N-tile n0 = (blk*4+w)*16, both batch halves.
// WMMA f32 16x16x4: A lane(l): M=l&15, K=k0+2*(l>>4)+vgpr ; B lane(l): N=l&15, same K ;
// C/D lane(l),vgpr v: M = v + 8*(l>>4), N = l&15.
__global__ void __launch_bounds__(128)
rnn_step(const float* __restrict__ h_prev,   // [32][1024]
         float* __restrict__ h_next,         // [32][1024]
         const float* __restrict__ W4,       // [k0/4][n][4]
         const float* __restrict__ Win,      // [i][n]
         const float* __restrict__ bias,     // [n]
         const int* __restrict__ x,          // [B][T]
         float* __restrict__ out,            // [B][T][256]
         int t) {
    __shared__ float hs[RNN_B * HS_STRIDE];  // 131584 B: padded H_prev stage

    const int tid  = threadIdx.x;
    const int lane = tid & 31;
    const int wave = tid >> 5;
    const int half = lane >> 4;   // K sub-block / batch sub-block selector
    const int lm   = lane & 15;
    const int n    = (blockIdx.x * 4 + wave) * 16 + lm;

    // cooperative stage of H_prev into padded LDS (float4 granularity)
    {
        const float4* src = (const float4*)h_prev;
        for (int i4 = tid; i4 < (RNN_B * RNN_N) / 4; i4 += 128) {
            int m = i4 >> 8;          // /256 float4 per row
            int j4 = i4 & 255;
            ((float4*)(hs + m * HS_STRIDE))[j4] = src[i4];
        }
    }
    __syncthreads();

    v8f c0 = {};   // batches 0..15
    v8f c1 = {};   // batches 16..31

    const float* hlo = hs + lm * HS_STRIDE;
    const float* hhi = hs + (lm + 16) * HS_STRIDE;

    for (int k0 = 0; k0 < RNN_N; k0 += 4) {
        const int k = k0 + 2 * half;
        v2f a0 = *(const v2f*)(hlo + k);
        v2f a1 = *(const v2f*)(hhi + k);
        v2f b  = *(const v2f*)(W4 + (unsigned)(k0 >> 2) * (RNN_N * 4) + n * 4 + 2 * half);
        c0 = __builtin_amdgcn_wmma_f32_16x16x4_f32(false, a0, false, b, (short)0, c0, false, false);
        c1 = __builtin_amdgcn_wmma_f32_16x16x4_f32(false, a1, false, b, (short)0, c1, false, false);
    }

    const float bn  = bias[n];
    const bool lin  = (n >= RNN_N - 256);

#pragma unroll
    for (int v = 0; v < 8; ++v) {
        int m0 = v + 8 * half;   // C-fragment row -> batch index
        int m1 = m0 + 16;
        float p0 = c0[v] + Win[(unsigned)x[m0 * RNN_T + t] * RNN_N + n] + bn;
        float p1 = c1[v] + Win[(unsigned)x[m1 * RNN_T + t] * RNN_N + n] + bn;
        if (!lin) {
            p0 = fminf(fmaxf(p0, 0.f), RELU_CAP);
            p1 = fminf(fmaxf(p1, 0.f), RELU_CAP);
        }
        h_next[m0 * RNN_N + n] = p0;
        h_next[m1 * RNN_N + n] = p1;
        if (lin) {  // neurons 768..1023 are exactly the logits slice
            out[((unsigned)m0 * RNN_T + t) * 256 + (n - 768)] = p0;
            out[((unsigned)m1 * RNN_T + t) * 256 + (n - 768)] = p1;
        }
    }
}

extern "C" void kernel_launch(void* const* d_in, const int* in_sizes, int n_in,
                              void* d_out, int out_size, void* d_ws, size_t ws_size,
                              hipStream_t stream) {
    (void)in_sizes; (void)n_in; (void)out_size; (void)ws_size;
    const int*   x       = (const int*)d_in[0];     // (B,T)
    // d_in[1] = tau (argmax invariant), d_in[4] = cl (unused by reference)
    const float* logits  = (const float*)d_in[2];   // (17408,16)
    const float* grid    = (const float*)d_in[3];   // (16,)
    const int*   in_src  = (const int*)d_in[5];     // (8192,)
    const int*   rec_src = (const int*)d_in[7];     // (8192,)
    float* out = (float*)d_out;                     // (B,T,256)

    float* ws   = (float*)d_ws;
    float* W4   = ws + OFF_W4;
    float* Win  = ws + OFF_WIN;
    float* bias = ws + OFF_BIAS;
    float* wall = ws + OFF_WALL;
    float* h0   = ws + OFF_H0;
    float* h1   = ws + OFF_H1;

    // 1) zero tables + initial hidden state (runs every call: deterministic)
    rnn_zero_ws<<<1362, 256, 0, stream>>>((float4*)d_ws, WS_FLOATS / 4);
    // 2) straight-through forward == grid[argmax]
    rnn_argmax_weights<<<(N_TOTAL + 255) / 256, 256, 0, stream>>>(logits, grid, wall);
    // 3) densify the sparse connectivity
    rnn_build_tables<<<RNN_N / 256, 256, 0, stream>>>(in_src, rec_src, wall, W4, Win, bias);
    // 4) sequential scan: one WMMA GEMM kernel per timestep (kernel boundary = step sync)
    float* hp = h0;
    float* hn = h1;
    for (int t = 0; t < RNN_T; ++t) {
        rnn_step<<<16, 128, 0, stream>>>(hp, hn, W4, Win, bias, x, out, t);
        float* tmp = hp; hp = hn; hn = tmp;
    }
}